// EssRegressor_23940147708044
// MI455X (gfx1250) — compile-verified
//
#include <hip/hip_runtime.h>

typedef __attribute__((ext_vector_type(2))) float v2f;
typedef __attribute__((ext_vector_type(4))) float v4f;
typedef __attribute__((ext_vector_type(8))) float v8f;

#define BATCH      32
#define IMG_H      480
#define IMG_W      640
#define NSAMP      10000
#define NGROUPS    313        // ceil(10000/32)
#define NWAVES     8
#define BLOCK_DIM  (NWAVES * 32)

__global__ __launch_bounds__(BLOCK_DIM)
void ess_gram_eig_kernel(const float* __restrict__ flow,   // [B,3,H,W]
                         const float* __restrict__ Kq,     // [B,3,3]
                         const float* __restrict__ Ks,     // [B,3,3]
                         const int*   __restrict__ idx,    // [NSAMP]
                         float*       __restrict__ out)    // [B,3,3]
{
    const int b    = blockIdx.x;
    const int tid  = threadIdx.x;
    const int lane = tid & 31;
    // Force wave index into an SGPR -> scalar loop, EXEC stays all-1s (WMMA req)
    const int wave = __builtin_amdgcn_readfirstlane(tid >> 5);

    __shared__ float tup[NWAVES][32][8];       // wave-private tuple staging (32B stride)
    __shared__ float partial[NWAVES][32][8];   // per-wave WMMA accumulators
    __shared__ float Amat[16][16];             // reduced Gram (9x9 used)
    __shared__ float Vm[9][9];                 // eigenvectors

    // Per-batch intrinsics (uniform -> scalar loads)
    const float fq  = Kq[b * 9 + 0];
    const float cxq = Kq[b * 9 + 2];
    const float cyq = Kq[b * 9 + 5];
    const float fs  = Ks[b * 9 + 0];
    const float cxs = Ks[b * 9 + 2];
    const float cys = Ks[b * 9 + 5];
    const float ifq = 1.0f / fq;
    const float ifs = 1.0f / fs;

    const size_t fbase = (size_t)b * 3u * IMG_H * IMG_W;
    const float* __restrict__ fl0 = flow + fbase;
    const float* __restrict__ fl1 = flow + fbase + (size_t)IMG_H * IMG_W;
    const float* __restrict__ fl2 = flow + fbase + (size_t)2 * IMG_H * IMG_W;

    // Lane feature index m = lane%16 -> (fi,fj); feature = p*q with
    // p = c0*x0 + c1*x1 + c2 (0 for padding rows m>8), q = d0*x2 + d1*x3 + d2
    const int m  = lane & 15;
    const int fi = m / 3;
    const int fj = m - 3 * fi;
    const float c0 = (fi == 0) ? 1.0f : 0.0f;
    const float c1 = (fi == 1) ? 1.0f : 0.0f;
    const float c2 = (fi == 2) ? 1.0f : 0.0f;
    const float d0 = (fj == 0) ? 1.0f : 0.0f;
    const float d1 = (fj == 1) ? 1.0f : 0.0f;
    const float d2 = (fj == 2) ? 1.0f : 0.0f;
    const int   k0 = (lane >= 16) ? 2 : 0;     // K index pair for this lane half

    v8f acc = {};  // 16x16 f32 WMMA accumulator (9x9 region meaningful)

    for (int g = wave; g < NGROUPS; g += NWAVES) {
        const int s  = g * 32 + lane;
        const int sc = (s < NSAMP) ? s : (NSAMP - 1);   // clamped: loads stay in-bounds

        const int   iv   = idx[sc];
        const int   vrow = iv / IMG_W;
        const int   ucol = iv - vrow * IMG_W;
        const float uf   = (float)ucol;
        const float vf   = (float)vrow;
        const float du   = fl0[iv];
        const float dv   = fl1[iv];
        const float wraw = fl2[iv];
        const float w    = (s < NSAMP) ? wraw : 0.0f;   // w=0 => zero contribution

        // normalized source (flowed, K_s) and query (grid, K_q) coordinates
        tup[wave][lane][0] = (uf + du - cxs) * ifs;   // x0
        tup[wave][lane][1] = (vf + dv - cys) * ifs;   // x1
        tup[wave][lane][2] = (uf - cxq) * ifq;        // x2
        tup[wave][lane][3] = (vf - cyq) * ifq;        // x3
        tup[wave][lane][4] = w;

        // 8 WMMA chunks of 4 samples; tuples re-read broadcast-style from LDS.
        // Per-wave DS ordering guarantees store->load visibility (no barrier).
        #pragma unroll
        for (int cc = 0; cc < 8; ++cc) {
            const int sl0 = cc * 4 + k0;
            const int sl1 = sl0 + 1;

            const v4f  t0 = *(const v4f*)&tup[wave][sl0][0];
            const float w0 = tup[wave][sl0][4];
            const v4f  t1 = *(const v4f*)&tup[wave][sl1][0];
            const float w1 = tup[wave][sl1][4];

            const float p0 = fmaf(c0, t0.x, fmaf(c1, t0.y, c2));
            const float q0 = fmaf(d0, t0.z, fmaf(d1, t0.w, d2));
            const float p1 = fmaf(c0, t1.x, fmaf(c1, t1.y, c2));
            const float q1 = fmaf(d0, t1.z, fmaf(d1, t1.w, d2));

            v2f a, bm;
            a.x  = p0 * q0;   a.y  = p1 * q1;
            bm.x = w0 * a.x;  bm.y = w1 * a.y;   // B = w .* A (m == n == lane%16)

            // D(16x16,f32) += A(16x4,f32) * B(4x16,f32)
            acc = __builtin_amdgcn_wmma_f32_16x16x4_f32(
                false, a, false, bm, (short)0, acc, false, false);
        }
    }

    // ---- cross-wave reduction of the 16x16 accumulator through LDS ----
    #pragma unroll
    for (int jj = 0; jj < 8; ++jj) partial[wave][lane][jj] = acc[jj];
    __syncthreads();

    if (wave == 0) {
        const int mbase = (lane < 16) ? 0 : 8;
        const int ncol  = lane & 15;
        #pragma unroll
        for (int jj = 0; jj < 8; ++jj) {
            float s2 = 0.0f;
            #pragma unroll
            for (int w2 = 0; w2 < NWAVES; ++w2) s2 += partial[w2][lane][jj];
            // C/D layout: VGPR jj -> M = jj (lanes 0-15) or 8+jj (lanes 16-31)
            Amat[mbase + jj][ncol] = s2;
        }
    }
    __syncthreads();

    // ---- 9x9 symmetric eigensolve (cyclic Jacobi), single thread ----
    if (tid == 0) {
        for (int r = 0; r < 9; ++r)
            for (int c = 0; c < 9; ++c)
                Vm[r][c] = (r == c) ? 1.0f : 0.0f;

        for (int sweep = 0; sweep < 10; ++sweep) {
            for (int p = 0; p < 8; ++p) {
                for (int q = p + 1; q < 9; ++q) {
                    const float apq = Amat[p][q];
                    if (fabsf(apq) < 1e-12f) continue;
                    const float app = Amat[p][p];
                    const float aqq = Amat[q][q];
                    const float tau = (aqq - app) / (2.0f * apq);
                    const float t   = ((tau >= 0.0f) ? 1.0f : -1.0f) /
                                      (fabsf(tau) + sqrtf(1.0f + tau * tau));
                    const float c   = 1.0f / sqrtf(1.0f + t * t);
                    const float sn  = t * c;

                    for (int r = 0; r < 9; ++r) {           // A <- A * J
                        const float arp = Amat[r][p];
                        const float arq = Amat[r][q];
                        Amat[r][p] = c * arp - sn * arq;
                        Amat[r][q] = sn * arp + c * arq;
                    }
                    for (int cc = 0; cc < 9; ++cc) {        // A <- J^T * A
                        const float apr = Amat[p][cc];
                        const float aqr = Amat[q][cc];
                        Amat[p][cc] = c * apr - sn * aqr;
                        Amat[q][cc] = sn * apr + c * aqr;
                    }
                    for (int r = 0; r < 9; ++r) {           // V <- V * J
                        const float vrp = Vm[r][p];
                        const float vrq = Vm[r][q];
                        Vm[r][p] = c * vrp - sn * vrq;
                        Vm[r][q] = sn * vrp + c * vrq;
                    }
                }
            }
        }

        // sort eigenpairs ascending by eigenvalue (diag of Amat)
        int ord[9];
        for (int k = 0; k < 9; ++k) ord[k] = k;
        for (int a2 = 0; a2 < 8; ++a2)
            for (int b2 = a2 + 1; b2 < 9; ++b2)
                if (Amat[ord[b2]][ord[b2]] < Amat[ord[a2]][ord[a2]]) {
                    const int tswap = ord[a2]; ord[a2] = ord[b2]; ord[b2] = tswap;
                }

        // reference takes eig_vectors[:, 0] (row 0 across sorted eigvecs)
        for (int k = 0; k < 9; ++k)
            out[b * 9 + k] = Vm[0][ord[k]];
    }
}

extern "C" void kernel_launch(void* const* d_in, const int* in_sizes, int n_in,
                              void* d_out, int out_size, void* d_ws, size_t ws_size,
                              hipStream_t stream) {
    const float* flow = (const float*)d_in[0];
    const float* Kq   = (const float*)d_in[1];
    const float* Ks   = (const float*)d_in[2];
    const int*   idx  = (const int*)d_in[3];
    float*       out  = (float*)d_out;

    ess_gram_eig_kernel<<<BATCH, BLOCK_DIM, 0, stream>>>(flow, Kq, Ks, idx, out);
}